// CenterWoParamMultiCosineLossV2_15917148799630
// MI455X (gfx1250) — compile-verified
//
#include <hip/hip_runtime.h>

#define Bn 8192
#define Dn 512
#define NC 90
#define Kc 8
#define NTILES 608      // >= 512 + 90 worst-case class-pure tiles
#define NCHUNK 4
#define CHUNK_D 128     // Dn / NCHUNK
#define STEPS 32        // CHUNK_D / 4
#define ROWSTRIDE 136   // CHUNK_D + 8 ; 136 mod 64 == 8 -> conflict-free B reads

// ws int-region layout (int indices into (int*)d_ws)
#define OFF_COUNTS 0      // 90
#define OFF_BASE   96     // 91
#define OFF_CURSOR 192    // 90
#define OFF_TCLS   288    // 608
#define OFF_TSTART 896    // 608
#define OFF_TCNT   1504   // 608
#define OFF_ORDER  2112   // 8192
#define PART_BYTE_OFF 49152           // float part[NCHUNK][Bn][Kc]
#define PARTIAL_BYTE_OFF (49152 + NCHUNK*Bn*Kc*4)

typedef float v2f __attribute__((ext_vector_type(2)));
typedef float v8f __attribute__((ext_vector_type(8)));

__global__ void k_zero(int* W) {
    int i = threadIdx.x;
    if (i < NC) W[OFF_COUNTS + i] = 0;
}

__global__ void k_hist(const int* __restrict__ labels, int* __restrict__ W) {
    int i = blockIdx.x * blockDim.x + threadIdx.x;
    if (i < Bn) atomicAdd(&W[OFF_COUNTS + labels[i]], 1);
}

__global__ void k_plan(int* W) {
    if (threadIdx.x != 0 || blockIdx.x != 0) return;
    int acc = 0;
    for (int c = 0; c < NC; ++c) {
        W[OFF_BASE + c] = acc;
        W[OFF_CURSOR + c] = acc;
        acc += W[OFF_COUNTS + c];
    }
    W[OFF_BASE + NC] = acc;
    int idx = 0;
    for (int c = 0; c < NC; ++c) {
        int cnt = W[OFF_COUNTS + c];
        for (int t = 0; t * 16 < cnt; ++t) {
            W[OFF_TCLS + idx] = c;
            W[OFF_TSTART + idx] = W[OFF_BASE + c] + t * 16;
            int r = cnt - t * 16;
            W[OFF_TCNT + idx] = r < 16 ? r : 16;
            ++idx;
        }
    }
    for (; idx < NTILES; ++idx) {
        W[OFF_TCLS + idx] = 0; W[OFF_TSTART + idx] = 0; W[OFF_TCNT + idx] = 0;
    }
}

__global__ void k_scatter(const int* __restrict__ labels, int* __restrict__ W) {
    int i = blockIdx.x * blockDim.x + threadIdx.x;
    if (i < Bn) {
        int pos = atomicAdd(&W[OFF_CURSOR + labels[i]], 1);
        W[OFF_ORDER + pos] = i;
    }
}

// One wave per workgroup; tile = class-pure group of up to 16 samples.
// D = A(16 x 4, gathered x rows) * B(4 x 16, class centers from LDS) + C,
// iterated 32x over this K-chunk with V_WMMA_F32_16X16X4_F32.
__global__ void __launch_bounds__(32)
k_main(const float* __restrict__ x, const float* __restrict__ centers,
       const int* __restrict__ W, float* __restrict__ part) {
    __shared__ float smem[Kc * ROWSTRIDE];
    const int tile  = blockIdx.x;
    const int chunk = blockIdx.y;
    const int cnt   = W[OFF_TCNT + tile];
    if (cnt <= 0) return;                       // wave-uniform
    const int cls   = W[OFF_TCLS + tile];
    const int start = W[OFF_TSTART + tile];
    const int* order = W + OFF_ORDER;
    const int lane = threadIdx.x;

    // Stage this class's center chunk: 8 rows x 128 floats, padded rows.
    {
        const float* src = centers + cls * Kc * Dn + chunk * CHUNK_D;
        #pragma unroll
        for (int t = 0; t < 8; ++t) {
            int idx = t * 128 + lane * 4;       // covers 0..1023 exactly
            int row = idx >> 7, col = idx & 127;
            float4 v = *(const float4*)(src + row * Dn + col);
            *(float4*)(&smem[row * ROWSTRIDE + col]) = v;
        }
    }
    __syncthreads();

    const int m  = lane & 15;                   // sample row in A
    const int kh = lane >> 4;                   // K-half (lanes 16-31 hold K=2,3)
    const int mm = m < cnt ? m : cnt - 1;       // clamp padding rows (results discarded)
    const float* xrow = x + order[start + mm] * Dn + chunk * CHUNK_D + 2 * kh;
    const float* brow = smem + (lane & 7) * ROWSTRIDE + 2 * kh;   // column n reads center row n&7

    v8f c0 = {}; v8f c1 = {};
    #pragma unroll
    for (int s = 0; s < STEPS; s += 2) {
        v2f a0 = *(const v2f*)(xrow + s * 4);
        v2f b0 = *(const v2f*)(brow + s * 4);
        c0 = __builtin_amdgcn_wmma_f32_16x16x4_f32(false, a0, false, b0,
                                                   (short)0, c0, false, false);
        v2f a1 = *(const v2f*)(xrow + s * 4 + 4);
        v2f b1 = *(const v2f*)(brow + s * 4 + 4);
        c1 = __builtin_amdgcn_wmma_f32_16x16x4_f32(false, a1, false, b1,
                                                   (short)0, c1, false, false);
    }

    // C/D layout: VGPR r -> M = r + 8*kh, N = lane&15. Keep columns n<8 (k index).
    const int n = lane & 15;
    if (n < Kc) {
        #pragma unroll
        for (int r = 0; r < 8; ++r) {
            int mi = r + 8 * kh;
            if (mi < cnt) {
                int b = order[start + mi];
                part[(chunk * Bn + b) * Kc + n] = c0[r] + c1[r];
            }
        }
    }
}

__global__ void k_finalize(const float* __restrict__ part, float* __restrict__ partials) {
    __shared__ float red[256];
    int b = blockIdx.x * 256 + threadIdx.x;
    float t = 0.f;
    {
        float den = 0.f, num = 0.f;
        #pragma unroll
        for (int k = 0; k < Kc; ++k) {
            float d = 1.0f;
            #pragma unroll
            for (int ch = 0; ch < NCHUNK; ++ch)
                d += part[(ch * Bn + b) * Kc + k];
            den += d;
            num += d * d;
        }
        t = num / den;
    }
    red[threadIdx.x] = t;
    __syncthreads();
    for (int s = 128; s > 0; s >>= 1) {
        if (threadIdx.x < s) red[threadIdx.x] += red[threadIdx.x + s];
        __syncthreads();
    }
    if (threadIdx.x == 0) partials[blockIdx.x] = red[0];
}

__global__ void k_final(const float* __restrict__ partials, float* __restrict__ out) {
    if (threadIdx.x == 0 && blockIdx.x == 0) {
        float s = 0.f;
        for (int i = 0; i < Bn / 256; ++i) s += partials[i];
        out[0] = s / (float)Bn;
    }
}

extern "C" void kernel_launch(void* const* d_in, const int* in_sizes, int n_in,
                              void* d_out, int out_size, void* d_ws, size_t ws_size,
                              hipStream_t stream) {
    const float* x       = (const float*)d_in[0];
    const float* centers = (const float*)d_in[1];
    const int*   labels  = (const int*)d_in[2];
    int*   W        = (int*)d_ws;
    float* part     = (float*)((char*)d_ws + PART_BYTE_OFF);
    float* partials = (float*)((char*)d_ws + PARTIAL_BYTE_OFF);

    k_zero    <<<1, 128, 0, stream>>>(W);
    k_hist    <<<Bn / 256, 256, 0, stream>>>(labels, W);
    k_plan    <<<1, 32, 0, stream>>>(W);
    k_scatter <<<Bn / 256, 256, 0, stream>>>(labels, W);
    k_main    <<<dim3(NTILES, NCHUNK), 32, 0, stream>>>(x, centers, W, part);
    k_finalize<<<Bn / 256, 256, 0, stream>>>(part, partials);
    k_final   <<<1, 32, 0, stream>>>(partials, (float*)d_out);
}